// AVWGCNT_41790031790514
// MI455X (gfx1250) — compile-verified
//
#include <hip/hip_runtime.h>
#include <hip/hip_bf16.h>

// Problem constants (match reference)
#define N_NODES 4096
#define EMBED   16
#define DIN     32
#define DOUT    64
#define CHEB_K  3
#define BATCH   16
#define LN_EPS  1e-12f
#define KI      (CHEB_K * DIN)   // 96
#define ROW_CHUNK 256            // rows per partial-softmax chunk
#define N_CHUNKS  (N_NODES / ROW_CHUNK)  // 16

typedef __attribute__((ext_vector_type(2))) float v2f;
typedef __attribute__((ext_vector_type(8))) float v8f;

// Branchless elu (alpha=1): max(x,0) + (exp(min(x,0)) - 1) is exact for both
// signs and keeps EXEC untouched (no divergent branches near WMMA regions).
__device__ __forceinline__ float elu1(float x) {
    return fmaxf(x, 0.0f) + (expf(fminf(x, 0.0f)) - 1.0f);
}

// ---------------------------------------------------------------------------
// 1) LayerNorm over embed dim + bias = e @ bias_pool
// ---------------------------------------------------------------------------
__global__ void k_ln_bias(const float* __restrict__ emb,
                          const float* __restrict__ bias_pool,
                          const float* __restrict__ gamma,
                          const float* __restrict__ beta,
                          float* __restrict__ e,
                          float* __restrict__ bias) {
    int n = blockIdx.x * blockDim.x + threadIdx.x;
    if (n >= N_NODES) return;
    float v[EMBED];
    float mu = 0.0f;
    #pragma unroll
    for (int d = 0; d < EMBED; ++d) { v[d] = emb[n * EMBED + d]; mu += v[d]; }
    mu *= (1.0f / EMBED);
    float var = 0.0f;
    #pragma unroll
    for (int d = 0; d < EMBED; ++d) { float t = v[d] - mu; var += t * t; }
    var *= (1.0f / EMBED);
    float inv = rsqrtf(var + LN_EPS);
    #pragma unroll
    for (int d = 0; d < EMBED; ++d) {
        v[d] = (v[d] - mu) * inv * gamma[d] + beta[d];
        e[n * EMBED + d] = v[d];
    }
    for (int o = 0; o < DOUT; ++o) {
        float s = 0.0f;
        #pragma unroll
        for (int d = 0; d < EMBED; ++d) s += v[d] * bias_pool[d * DOUT + o];
        bias[n * DOUT + o] = s;
    }
}

// ---------------------------------------------------------------------------
// 2) A = elu(e @ e^T), one wave per 16x16 tile, fp32 WMMA, K=16 in 4 steps
// ---------------------------------------------------------------------------
__global__ void __launch_bounds__(128) k_gram_elu(const float* __restrict__ e,
                                                  float* __restrict__ S) {
    int wid  = blockIdx.x * 4 + (threadIdx.x >> 5);
    int lane = threadIdx.x & 31;
    int rt = wid >> 8, ct = wid & 255;
    int r0 = rt * 16, c0 = ct * 16;
    int half = lane >> 4;          // 0: K pair {0,1}, 1: K pair {2,3}
    int l15  = lane & 15;
    int kk   = half * 2;
    v8f acc = {};
    #pragma unroll
    for (int k0 = 0; k0 < EMBED; k0 += 4) {
        v2f a = *(const v2f*)(e + (r0 + l15) * EMBED + k0 + kk);  // A[m, k..k+1]
        v2f b = *(const v2f*)(e + (c0 + l15) * EMBED + k0 + kk);  // B[k..k+1, n] = e[n, k]
        acc = __builtin_amdgcn_wmma_f32_16x16x4_f32(
                  false, a, false, b, (short)0, acc, false, false);
    }
    #pragma unroll
    for (int v = 0; v < 8; ++v) {
        int row = v + half * 8;
        S[(size_t)(r0 + row) * N_NODES + c0 + l15] = elu1(acc[v]);
    }
}

// ---------------------------------------------------------------------------
// 3a) Partial column-softmax stats: 16 row chunks x 4096 columns.
//     Branchless online (max, rescaled sum); coalesced row sweeps.
// ---------------------------------------------------------------------------
__global__ void k_col_stats_partial(const float* __restrict__ S,
                                    float* __restrict__ pmax,
                                    float* __restrict__ psum) {
    int m     = blockIdx.x * blockDim.x + threadIdx.x;  // column
    int chunk = blockIdx.y;
    int nbeg  = chunk * ROW_CHUNK;
    float mx = -3.402823466e38f;
    float sum = 0.0f;
    for (int n = nbeg; n < nbeg + ROW_CHUNK; ++n) {
        float v  = S[(size_t)n * N_NODES + m];
        float nm = fmaxf(mx, v);
        sum = sum * expf(mx - nm) + expf(v - nm);
        mx  = nm;
    }
    pmax[chunk * N_NODES + m] = mx;
    psum[chunk * N_NODES + m] = sum;
}

// ---------------------------------------------------------------------------
// 3b) Combine the 16 partials per column: sum_i s_i * exp(mx_i - mx_total)
// ---------------------------------------------------------------------------
__global__ void k_col_stats_combine(const float* __restrict__ pmax,
                                    const float* __restrict__ psum,
                                    float* __restrict__ colmax,
                                    float* __restrict__ colsum) {
    int m = blockIdx.x * blockDim.x + threadIdx.x;
    float mx = -3.402823466e38f;
    #pragma unroll
    for (int c = 0; c < N_CHUNKS; ++c) mx = fmaxf(mx, pmax[c * N_NODES + m]);
    float sum = 0.0f;
    #pragma unroll
    for (int c = 0; c < N_CHUNKS; ++c)
        sum += psum[c * N_NODES + m] * expf(pmax[c * N_NODES + m] - mx);
    colmax[m] = mx;
    colsum[m] = sum;
}

// ---------------------------------------------------------------------------
// 4) S = exp(A - colmax) / colsum  (in place, elementwise)
// ---------------------------------------------------------------------------
__global__ void k_col_normalize(float* __restrict__ S,
                                const float* __restrict__ colmax,
                                const float* __restrict__ colsum) {
    size_t idx = (size_t)blockIdx.x * blockDim.x + threadIdx.x;
    int m = (int)(idx & (N_NODES - 1));
    S[idx] = expf(S[idx] - colmax[m]) / colsum[m];
}

// ---------------------------------------------------------------------------
// 5) Y[b] = S @ X[b] for all 16 batches. One wave computes a 16x32 output
//    block (both DIN c-tiles), reusing each S fragment for two WMMAs.
//    K = 4096 stepped by 4 with fp32 WMMA; prefetch ahead on the S stream.
// ---------------------------------------------------------------------------
__global__ void __launch_bounds__(128) k_prop(const float* __restrict__ S,
                                              const float* __restrict__ X,
                                              float* __restrict__ Y) {
    int wid  = blockIdx.x * 4 + (threadIdx.x >> 5);   // 4096 waves total
    int lane = threadIdx.x & 31;
    int b    = wid >> 8;                 // 16 batches
    int nt   = wid & 255;                // 256 row tiles
    int n0 = nt * 16;
    int half = lane >> 4, l15 = lane & 15;
    int kk = half * 2;
    const float* Xb   = X + (size_t)b * N_NODES * DIN;
    const float* Srow = S + (size_t)(n0 + l15) * N_NODES + kk;
    v8f acc0 = {}, acc1 = {};
    #pragma unroll 4
    for (int k0 = 0; k0 < N_NODES; k0 += 4) {
        if ((k0 & 255) == 0) __builtin_prefetch(Srow + k0 + 512, 0, 0);
        v2f a = *(const v2f*)(Srow + k0);                 // S[n, k..k+1]
        const float* xr0 = Xb + (size_t)(k0 + kk) * DIN + l15;
        v2f b0, b1;
        b0.x = xr0[0];        // X[k,   c]      (c-tile 0)
        b0.y = xr0[DIN];      // X[k+1, c]
        b1.x = xr0[16];       // X[k,   c+16]   (c-tile 1)
        b1.y = xr0[DIN + 16]; // X[k+1, c+16]
        acc0 = __builtin_amdgcn_wmma_f32_16x16x4_f32(
                   false, a, false, b0, (short)0, acc0, false, false);
        acc1 = __builtin_amdgcn_wmma_f32_16x16x4_f32(
                   false, a, false, b1, (short)0, acc1, false, false);
    }
    float* Yb = Y + (size_t)b * N_NODES * DIN;
    #pragma unroll
    for (int v = 0; v < 8; ++v) {
        int row = v + half * 8;
        Yb[(size_t)(n0 + row) * DIN + l15]      = acc0[v];
        Yb[(size_t)(n0 + row) * DIN + 16 + l15] = acc1[v];
    }
}

// ---------------------------------------------------------------------------
// 6) Per-node gconv: build W_n = sum_d e[n,d]*Wpool[d] (96x64) and gathered
//    x_g[b, 96] (with T2·x = 2*y2 - x folded in) in LDS, then 4 waves compute
//    the [B=16, DOUT=64] output via WMMA fed by ds loads. out += bias[n].
// ---------------------------------------------------------------------------
__global__ void __launch_bounds__(128) k_gconv(const float* __restrict__ e,
                                               const float* __restrict__ Wpool,
                                               const float* __restrict__ bias,
                                               const float* __restrict__ X,
                                               const float* __restrict__ Y1,
                                               const float* __restrict__ Y2,
                                               float* __restrict__ out) {
    __shared__ float sW[KI * DOUT];      // 96 x 64 = 24 KB
    __shared__ float sXg[BATCH * KI];    // 16 x 96 =  6 KB
    __shared__ float se[EMBED];
    int n   = blockIdx.x;
    int tid = threadIdx.x;
    if (tid < EMBED) se[tid] = e[n * EMBED + tid];
    __syncthreads();
    // Per-node weights from the hypernetwork pool (coalesced over idx).
    for (int idx = tid; idx < KI * DOUT; idx += 128) {
        float s = 0.0f;
        #pragma unroll
        for (int d = 0; d < EMBED; ++d)
            s += se[d] * Wpool[d * (KI * DOUT) + idx];
        sW[idx] = s;
    }
    // Gather x_g[b, ki]: T0 x = x, T1 x = y1, T2 x = 2*y2 - x.
    for (int idx = tid; idx < BATCH * KI; idx += 128) {
        int b  = idx / KI;
        int ki = idx % KI;
        int k = ki >> 5, i = ki & 31;
        size_t p = (size_t)b * N_NODES * DIN + (size_t)n * DIN + i;
        float v;
        if (k == 0)      v = X[p];
        else if (k == 1) v = Y1[p];
        else             v = 2.0f * Y2[p] - X[p];
        sXg[idx] = v;
    }
    __syncthreads();
    int wave = tid >> 5, lane = tid & 31;
    int half = lane >> 4, l15 = lane & 15;
    int kk = half * 2;
    int c0 = wave * 16;                  // 4 waves -> 4 output column tiles
    v8f acc = {};
    #pragma unroll
    for (int k0 = 0; k0 < KI; k0 += 4) {
        v2f a = *(const v2f*)(&sXg[l15 * KI + k0 + kk]);  // A[b, k..k+1]
        v2f bb;
        bb.x = sW[(k0 + kk    ) * DOUT + c0 + l15];
        bb.y = sW[(k0 + kk + 1) * DOUT + c0 + l15];
        acc = __builtin_amdgcn_wmma_f32_16x16x4_f32(
                  false, a, false, bb, (short)0, acc, false, false);
    }
    float bv = bias[n * DOUT + c0 + l15];
    #pragma unroll
    for (int v = 0; v < 8; ++v) {
        int brow = v + half * 8;         // batch index
        out[(size_t)brow * N_NODES * DOUT + (size_t)n * DOUT + c0 + l15] = acc[v] + bv;
    }
}

// ---------------------------------------------------------------------------
// Launch
// ---------------------------------------------------------------------------
extern "C" void kernel_launch(void* const* d_in, const int* in_sizes, int n_in,
                              void* d_out, int out_size, void* d_ws, size_t ws_size,
                              hipStream_t stream) {
    const float* x     = (const float*)d_in[0];   // [16, 4096, 32]
    const float* emb   = (const float*)d_in[1];   // [4096, 16]
    const float* wpool = (const float*)d_in[2];   // [16, 3, 32, 64]
    const float* bpool = (const float*)d_in[3];   // [16, 64]
    const float* gamma = (const float*)d_in[4];   // [16]
    const float* beta  = (const float*)d_in[5];   // [16]
    float* out = (float*)d_out;                   // [16, 4096, 64]
    float* ws  = (float*)d_ws;

    // Workspace layout (floats): ~85.8 MB total
    float* e      = ws;                       // 4096*16      =    65536
    float* bias   = e + 65536;                // 4096*64      =   262144
    float* colmax = bias + 262144;            // 4096
    float* colsum = colmax + 4096;            // 4096
    float* pmax   = colsum + 4096;            // 16*4096      =    65536
    float* psum   = pmax + 65536;             // 16*4096      =    65536
    float* y1     = psum + 65536;             // 16*4096*32   =  2097152
    float* y2     = y1 + 2097152;             // 16*4096*32   =  2097152
    float* S      = y2 + 2097152;             // 4096*4096    = 16777216

    k_ln_bias<<<16, 256, 0, stream>>>(emb, bpool, gamma, beta, e, bias);
    k_gram_elu<<<16384, 128, 0, stream>>>(e, S);
    k_col_stats_partial<<<dim3(16, N_CHUNKS), 256, 0, stream>>>(S, pmax, psum);
    k_col_stats_combine<<<16, 256, 0, stream>>>(pmax, psum, colmax, colsum);
    k_col_normalize<<<65536, 256, 0, stream>>>(S, colmax, colsum);
    k_prop<<<1024, 128, 0, stream>>>(S, x,  y1);   // y1 = S @ x
    k_prop<<<1024, 128, 0, stream>>>(S, y1, y2);   // y2 = S @ y1
    k_gconv<<<4096, 128, 0, stream>>>(e, wpool, bias, x, y1, y2, out);
}